// Coo2BookKeeping_28278064676905
// MI455X (gfx1250) — compile-verified
//
#include <hip/hip_runtime.h>

// CDNA5 / gfx1250 — edge bookkeeping kernel (coo2_adj_vec_sod).
// Memory-bound: ~52 B/edge * 2.56M edges ~ 133 MB -> ~6 us @ 23.3 TB/s.
// Strategy:
//   * Precompute all B*NS Cartesian shift vectors ONCE per block via
//     V_WMMA_F32_16X16X4_F32 (the only matrix math in the workload) into LDS.
//   * Stream adj/sft with non-temporal loads, outputs with non-temporal
//     stores (read/written exactly once; don't pollute L2).
//   * Gather positions (480 KB, ~128x reuse) through the cache hierarchy.

typedef __attribute__((ext_vector_type(2))) float v2f;
typedef __attribute__((ext_vector_type(8))) float v8f;

#define TPB 256          // 8 wave32 per block
#define EPT 4            // edges per thread (MLP)
#define TAB_STRIDE 17    // padded LDS row stride (dwords) -> fewer bank conflicts

__global__ __launch_bounds__(TPB) void coo2_edges_kernel(
    const int*   __restrict__ adj0,   // [E] source atom (i)
    const int*   __restrict__ adj1,   // [E] dest atom (j)
    const int*   __restrict__ sft,    // [E] shift-table index
    const float* __restrict__ spc,    // [NS][3] fractional shifts
    const float* __restrict__ pos,    // [B*N][3] positions
    const float* __restrict__ cel,    // [B][3][3] cell matrices
    float*       __restrict__ vec_out,// [E][3]
    float*       __restrict__ sod_out,// [E]
    int E, int N, int B, int NS, float rc2)
{
    // tab[s][b*3+k] = sum_d spc[s][d] * cel[b][d][k]   (Cartesian shift)
    __shared__ float tab[32 * TAB_STRIDE];

    // ---------------- Stage 1: shift table via the matrix unit -------------
    // D(16x16) = A(16x4) x B(4x16):
    //   A[s][d] = spc[s][d]            (d<3, else 0; two row-tiles for s=0..31)
    //   B[d][n] = cel[n/3][d][n%3]     (n = b*3+k packs all batches in columns)
    // => D[s][b*3+k] = shift_cart[b][s][k]
    const int lane = threadIdx.x & 31;
    const int wave = threadIdx.x >> 5;
    if (wave == 0) {
        const int half = lane >> 4;   // 0 -> K=0,1 ; 1 -> K=2,3
        const int m    = lane & 15;
        const int k0   = half * 2;

        // B fragment (shared by both A tiles)
        const int bb = m / 3;
        const int kk = m - bb * 3;
        float fb0 = (bb < B && (k0    ) < 3) ? cel[bb * 9 + (k0    ) * 3 + kk] : 0.0f;
        float fb1 = (bb < B && (k0 + 1) < 3) ? cel[bb * 9 + (k0 + 1) * 3 + kk] : 0.0f;
        v2f fb; fb.x = fb0; fb.y = fb1;

        v8f c = {};
        #pragma unroll
        for (int tile = 0; tile < 2; ++tile) {
            const int s = tile * 16 + m;
            v2f fa;
            fa.x = (s < NS && (k0    ) < 3) ? spc[s * 3 + k0    ] : 0.0f;
            fa.y = (s < NS && (k0 + 1) < 3) ? spc[s * 3 + k0 + 1] : 0.0f;
            // 8 args: (neg_a, A, neg_b, B, c_mod, C, reuse_a, reuse_b)
            v8f d = __builtin_amdgcn_wmma_f32_16x16x4_f32(
                false, fa, false, fb, (short)0, c, false, false);
            // D layout: lanes 0-15 col=m rows 0..7; lanes 16-31 col=m rows 8..15
            #pragma unroll
            for (int r = 0; r < 8; ++r) {
                const int row = tile * 16 + half * 8 + r;
                tab[row * TAB_STRIDE + m] = d[r];
            }
        }
    }
    __syncthreads();

    // ---------------- Stage 2: stream the edges ----------------------------
    const int base = blockIdx.x * (TPB * EPT) + threadIdx.x;

    int   ia[EPT], ja[EPT], sa[EPT];
    float pix[EPT], piy[EPT], piz[EPT];
    float pjx[EPT], pjy[EPT], pjz[EPT];

    // Phase 1: non-temporal index loads (stream-once data)
    #pragma unroll
    for (int u = 0; u < EPT; ++u) {
        const int e = base + u * TPB;
        const bool ok = e < E;
        ia[u] = ok ? __builtin_nontemporal_load(adj0 + e) : 0;
        ja[u] = ok ? __builtin_nontemporal_load(adj1 + e) : 0;
        sa[u] = ok ? __builtin_nontemporal_load(sft  + e) : 0;
    }

    // Phase 2: cached position gathers (L2-resident, heavy reuse)
    #pragma unroll
    for (int u = 0; u < EPT; ++u) {
        const float* pi = pos + 3 * ia[u];
        const float* pj = pos + 3 * ja[u];
        pix[u] = pi[0]; piy[u] = pi[1]; piz[u] = pi[2];
        pjx[u] = pj[0]; pjy[u] = pj[1]; pjz[u] = pj[2];
    }

    // Phase 3: compute + non-temporal stores
    #pragma unroll
    for (int u = 0; u < EPT; ++u) {
        const int e = base + u * TPB;
        const unsigned b = (unsigned)ja[u] / (unsigned)N;   // batch of edge
        const float* t = &tab[sa[u] * TAB_STRIDE + (int)b * 3];
        float vx = pjx[u] - pix[u] + t[0];
        float vy = pjy[u] - piy[u] + t[1];
        float vz = pjz[u] - piz[u] + t[2];
        float sod = vx * vx + vy * vy + vz * vz;
        const bool keep = (sod < rc2) && (sod > 0.0f);
        vx  = keep ? vx  : 0.0f;
        vy  = keep ? vy  : 0.0f;
        vz  = keep ? vz  : 0.0f;
        sod = keep ? sod : 0.0f;
        if (e < E) {
            __builtin_nontemporal_store(vx,  vec_out + 3 * e + 0);
            __builtin_nontemporal_store(vy,  vec_out + 3 * e + 1);
            __builtin_nontemporal_store(vz,  vec_out + 3 * e + 2);
            __builtin_nontemporal_store(sod, sod_out + e);
        }
    }
}

extern "C" void kernel_launch(void* const* d_in, const int* in_sizes, int n_in,
                              void* d_out, int out_size, void* d_ws, size_t ws_size,
                              hipStream_t stream) {
    const int*   adj = (const int*)  d_in[0];   // [2,E]
    const int*   sft = (const int*)  d_in[1];   // [E]
    const float* spc = (const float*)d_in[2];   // [NS,3]
    const float* pos = (const float*)d_in[3];   // [B,N,3]
    const float* cel = (const float*)d_in[4];   // [B,3,3]

    const int E  = in_sizes[1];
    const int NS = in_sizes[2] / 3;
    const int B  = in_sizes[4] / 9;
    const int N  = (in_sizes[3] / 3) / B;

    float* vec_out = (float*)d_out;                  // first E*3 floats
    float* sod_out = (float*)d_out + (size_t)E * 3;  // next E floats

    const int tile   = TPB * EPT;
    const int blocks = (E + tile - 1) / tile;

    coo2_edges_kernel<<<blocks, TPB, 0, stream>>>(
        adj, adj + E, sft, spc, pos, cel,
        vec_out, sod_out, E, N, B, NS, 36.0f /* RC^2 = 6.0^2 */);
}